// OneHopSumNodeLabelAggregator_88235808129716
// MI455X (gfx1250) — compile-verified
//
#include <hip/hip_runtime.h>

// One-hop neighbor sum: out[d] += x[s] for each edge (s, d).
// N_NODES=100000, N_EDGES=1250000, D_FEAT=64 (f32).
//
// Strategy (MI455X): the naive scatter needs 80M f32 atomics -> L2 atomic-op
// throughput bound (>>100us), while the bandwidth roofline is ~3-7us.
// So we build a CSR-by-destination index with only 2.5M int atomics, then do
// an atomic-free, fully coalesced gather-reduce (b128 loads/stores), with the
// inner gather loop unrolled x2 for memory-level parallelism.

static constexpr int kNodes  = 100000;
static constexpr int kEdges  = 1250000;
static constexpr int kFeat   = 64;        // floats per node row
static constexpr int kChunks = kFeat / 4; // 16 float4 chunks per row
static constexpr int kScanB  = 1024;                       // scan block size
static constexpr int kNBlk   = (kNodes + kScanB - 1) / kScanB; // 98 partials

// ---------------- workspace layout (ints) ----------------
// cnt : [0, N)          per-dst edge counts
// off : [N, 2N+1)       exclusive offsets (off[N] = E)
// cur : [2N+1, 3N+1)    fill cursors
// blk : [3N+1, 3N+129)  scan partials (padded to 128)
// esrc: [3N+129, 3N+129+E)  binned source node ids

__device__ __forceinline__ int atom_add_i32(int* p, int v) {
  return __hip_atomic_fetch_add(p, v, __ATOMIC_RELAXED, __HIP_MEMORY_SCOPE_AGENT);
}

// ---- CSR path kernels -------------------------------------------------

__global__ __launch_bounds__(256) void zero_i32_kernel(int* __restrict__ p, int n) {
  int i = blockIdx.x * 256 + threadIdx.x;
  if (i < n) p[i] = 0;
}

__global__ __launch_bounds__(256) void count_kernel(const int* __restrict__ dst,
                                                    int* __restrict__ cnt, int nE) {
  int e = blockIdx.x * 256 + threadIdx.x;
  if (e < nE) atom_add_i32(&cnt[dst[e]], 1);   // global_atomic_add_u32 (no return)
}

// Hillis-Steele inclusive scan per 1024-block; writes block-exclusive prefix
// and the block total.
__global__ __launch_bounds__(kScanB) void scan_block_kernel(
    const int* __restrict__ cnt, int* __restrict__ off, int* __restrict__ blk, int n) {
  __shared__ int sm[kScanB];
  int t = threadIdx.x;
  int i = blockIdx.x * kScanB + t;
  int v = (i < n) ? cnt[i] : 0;
  sm[t] = v;
  __syncthreads();
  for (int d = 1; d < kScanB; d <<= 1) {
    int a = (t >= d) ? sm[t - d] : 0;
    __syncthreads();
    sm[t] += a;
    __syncthreads();
  }
  if (i < n) off[i] = sm[t] - v;              // exclusive within block
  if (t == kScanB - 1) blk[blockIdx.x] = sm[t];
}

// Single block scans the (<=128) block totals into exclusive block offsets.
__global__ __launch_bounds__(128) void scan_partials_kernel(int* __restrict__ blk, int nb) {
  __shared__ int sm[128];
  int t = threadIdx.x;
  int v = (t < nb) ? blk[t] : 0;
  sm[t] = v;
  __syncthreads();
  for (int d = 1; d < 128; d <<= 1) {
    int a = (t >= d) ? sm[t - d] : 0;
    __syncthreads();
    sm[t] += a;
    __syncthreads();
  }
  if (t < nb) blk[t] = sm[t] - v;             // exclusive
}

// off[i] += blk[i/1024]; cur[i] = off[i]; off[N] = E.
__global__ __launch_bounds__(256) void add_off_kernel(
    int* __restrict__ off, int* __restrict__ cur, const int* __restrict__ blk,
    int n, int eTotal) {
  int i = blockIdx.x * 256 + threadIdx.x;
  if (i < n) {
    int o = off[i] + blk[i >> 10];
    off[i] = o;
    cur[i] = o;
  }
  if (i == 0) off[n] = eTotal;
}

__global__ __launch_bounds__(256) void fill_kernel(
    const int* __restrict__ src, const int* __restrict__ dst,
    int* __restrict__ cur, int* __restrict__ esrc, int nE) {
  int e = blockIdx.x * 256 + threadIdx.x;
  if (e < nE) {
    int p = atom_add_i32(&cur[dst[e]], 1);    // returning global_atomic_add_u32
    esrc[p] = src[e];
  }
}

// 16 lanes (half a wave32) per node; lane c owns float4 chunk c of the row.
// All x loads / out stores are b128 and contiguous across the 16-lane group.
// Inner loop is unrolled x2 with independent accumulators so each iteration
// has two id-load -> row-load chains in flight.
__global__ __launch_bounds__(256) void gather_kernel(
    const float4* __restrict__ x, const int* __restrict__ off,
    const int* __restrict__ esrc, float4* __restrict__ out, int nNodes) {
  unsigned tid  = blockIdx.x * 256u + threadIdx.x;
  unsigned node = tid >> 4;
  unsigned c    = tid & 15u;
  if (node >= (unsigned)nNodes) return;

  int b = off[node];
  int e = off[node + 1];

  float4 acc0{0.0f, 0.0f, 0.0f, 0.0f};
  float4 acc1{0.0f, 0.0f, 0.0f, 0.0f};

  int k = b;
  for (; k + 1 < e; k += 2) {
    int s0 = esrc[k];
    int s1 = esrc[k + 1];
    float4 m0 = x[(unsigned)s0 * (unsigned)kChunks + c];  // global_load_b128
    float4 m1 = x[(unsigned)s1 * (unsigned)kChunks + c];  // global_load_b128
    acc0.x += m0.x; acc0.y += m0.y; acc0.z += m0.z; acc0.w += m0.w;
    acc1.x += m1.x; acc1.y += m1.y; acc1.z += m1.z; acc1.w += m1.w;
  }
  if (k < e) {
    int s = esrc[k];
    float4 m = x[(unsigned)s * (unsigned)kChunks + c];
    acc0.x += m.x; acc0.y += m.y; acc0.z += m.z; acc0.w += m.w;
  }

  float4 r{acc0.x + acc1.x, acc0.y + acc1.y, acc0.z + acc1.z, acc0.w + acc1.w};
  out[node * (unsigned)kChunks + c] = r;                  // global_store_b128
}

// ---- fallback (small workspace): direct atomic scatter ----------------

__global__ __launch_bounds__(256) void zero_f4_kernel(float4* __restrict__ o, int n4) {
  int i = blockIdx.x * 256 + threadIdx.x;
  if (i < n4) o[i] = float4{0.0f, 0.0f, 0.0f, 0.0f};
}

__global__ __launch_bounds__(256) void scatter_sum_kernel(
    const float4* __restrict__ x, const int* __restrict__ src,
    const int* __restrict__ dst, float* __restrict__ out) {
  unsigned tid = blockIdx.x * 256u + threadIdx.x;
  unsigned e = tid >> 4;
  unsigned c = tid & 15u;
  if (e >= (unsigned)kEdges) return;
  int s = src[e];
  int d = dst[e];
  float4 m = x[(unsigned)s * (unsigned)kChunks + c];
  float* o = out + (size_t)d * kFeat + (size_t)c * 4u;
  __hip_atomic_fetch_add(o + 0, m.x, __ATOMIC_RELAXED, __HIP_MEMORY_SCOPE_AGENT);
  __hip_atomic_fetch_add(o + 1, m.y, __ATOMIC_RELAXED, __HIP_MEMORY_SCOPE_AGENT);
  __hip_atomic_fetch_add(o + 2, m.z, __ATOMIC_RELAXED, __HIP_MEMORY_SCOPE_AGENT);
  __hip_atomic_fetch_add(o + 3, m.w, __ATOMIC_RELAXED, __HIP_MEMORY_SCOPE_AGENT);
}

// ---- host launcher ----------------------------------------------------

extern "C" void kernel_launch(void* const* d_in, const int* in_sizes, int n_in,
                              void* d_out, int out_size, void* d_ws, size_t ws_size,
                              hipStream_t stream) {
  (void)in_sizes; (void)n_in; (void)out_size;

  const float* x    = (const float*)d_in[0];
  const int*   ei   = (const int*)d_in[1];  // [2, kEdges]: row 0 = src, row 1 = dst
  const int*   src  = ei;
  const int*   dstp = ei + kEdges;
  float*       out  = (float*)d_out;

  const size_t wsNeeded = (size_t)(3 * kNodes + 129 + kEdges) * sizeof(int);

  if (ws_size >= wsNeeded && d_ws != nullptr) {
    int* ws   = (int*)d_ws;
    int* cnt  = ws;
    int* off  = ws + kNodes;
    int* cur  = ws + 2 * kNodes + 1;
    int* blk  = ws + 3 * kNodes + 1;
    int* esrc = ws + 3 * kNodes + 129;

    zero_i32_kernel<<<(kNodes + 255) / 256, 256, 0, stream>>>(cnt, kNodes);
    count_kernel<<<(kEdges + 255) / 256, 256, 0, stream>>>(dstp, cnt, kEdges);
    scan_block_kernel<<<kNBlk, kScanB, 0, stream>>>(cnt, off, blk, kNodes);
    scan_partials_kernel<<<1, 128, 0, stream>>>(blk, kNBlk);
    add_off_kernel<<<(kNodes + 255) / 256, 256, 0, stream>>>(off, cur, blk, kNodes, kEdges);
    fill_kernel<<<(kEdges + 255) / 256, 256, 0, stream>>>(src, dstp, cur, esrc, kEdges);
    gather_kernel<<<(kNodes * 16 + 255) / 256, 256, 0, stream>>>(
        (const float4*)x, off, esrc, (float4*)out, kNodes);
  } else {
    // Fallback: direct f32 atomic scatter (80M global_atomic_add_f32).
    int n4 = kNodes * kFeat / 4;
    zero_f4_kernel<<<(n4 + 255) / 256, 256, 0, stream>>>((float4*)out, n4);
    unsigned total = (unsigned)kEdges * 16u;
    scatter_sum_kernel<<<(total + 255u) / 256u, 256, 0, stream>>>(
        (const float4*)x, src, dstp, out);
  }
}